// FEconvNet_14121852470124
// MI455X (gfx1250) — compile-verified
//
#include <hip/hip_runtime.h>

#define N 64
#define NEL   (N * N * N)   // 262144 elements
#define NDOF  (3 * NEL)     // 786432 dofs
#define NCHUNK (NEL / 16)   // 16384 chunks of 16 type-sorted elements

typedef __attribute__((ext_vector_type(2))) float v2f;
typedef __attribute__((ext_vector_type(8))) float v8f;
typedef __attribute__((ext_vector_type(4))) unsigned int u32x4;
typedef __attribute__((ext_vector_type(8))) int i32x8;
typedef __attribute__((ext_vector_type(4))) int i32x4;

// ---------------- workspace layout (32-bit words) ----------------
// [0, NEL)                : types[e]
// [NEL, 2*NEL)            : sorted element ids (grouped by type)
// [2*NEL, 2*NEL+256)      : per-type counts
// [2*NEL+256, 2*NEL+512)  : per-type cursors (exclusive prefix)

__global__ void k_init(float* __restrict__ out, int* __restrict__ counts) {
  int i = blockIdx.x * blockDim.x + threadIdx.x;
  if (i < NDOF) out[i] = 0.0f;
  if (i < 256) counts[i] = 0;
}

// type code: t = sum over corners (di,dj,dk) of 2^(di*4+dj*2+dk) * (rho>0.5)
__global__ void k_types(const float* __restrict__ rho, int* __restrict__ types,
                        int* __restrict__ counts) {
  int e = blockIdx.x * blockDim.x + threadIdx.x;
  if (e >= NEL) return;
  int i = e / (N * N), j = (e / N) % N, k = e % N;
  int t = 0;
#pragma unroll
  for (int di = 0; di < 2; ++di)
#pragma unroll
    for (int dj = 0; dj < 2; ++dj)
#pragma unroll
      for (int dk = 0; dk < 2; ++dk) {
        int ii = (i + di) & (N - 1), jj = (j + dj) & (N - 1), kk = (k + dk) & (N - 1);
        if (rho[(ii * N + jj) * N + kk] > 0.5f) t += 1 << (di * 4 + dj * 2 + dk);
      }
  types[e] = t;
  atomicAdd(&counts[t], 1);
}

__global__ void k_scan(const int* __restrict__ counts, int* __restrict__ cursors) {
  __shared__ int tmp[256];
  int t = threadIdx.x;
  int c = counts[t];
  tmp[t] = c;
  __syncthreads();
  for (int off = 1; off < 256; off <<= 1) {
    int v = (t >= off) ? tmp[t - off] : 0;
    __syncthreads();
    tmp[t] += v;
    __syncthreads();
  }
  cursors[t] = tmp[t] - c;  // exclusive prefix sum
}

__global__ void k_scatter(const int* __restrict__ types, int* __restrict__ cursors,
                          int* __restrict__ sorted) {
  int e = blockIdx.x * blockDim.x + threadIdx.x;
  if (e >= NEL) return;
  int pos = atomicAdd(&cursors[types[e]], 1);
  sorted[pos] = e;
}

// 12 WMMAs covering KUe[16x24] += (masked Ue[16x24]) x Ke[24x24].
// `ke` may point to LDS (TDM-staged) or global (fallback) — inlined both ways.
__device__ __forceinline__ void run_wmma_12(const float* ke, const float* ue,
                                            float amask, int mrow, int half,
                                            v8f& acc0, v8f& acc1) {
  int col0 = mrow;
  int col1 = 16 + mrow;
  float s1 = (col1 < 24) ? 1.0f : 0.0f;
  int   c1 = (col1 < 24) ? col1 : 0;
#pragma unroll
  for (int kc = 0; kc < 6; ++kc) {
    int j0 = kc * 4 + half * 2;
    v2f a;  a.x  = amask * ue[j0];
            a.y  = amask * ue[j0 + 1];
    v2f b0; b0.x = ke[j0 * 24 + col0];
            b0.y = ke[(j0 + 1) * 24 + col0];
    v2f b1; b1.x = s1 * ke[j0 * 24 + c1];
            b1.y = s1 * ke[(j0 + 1) * 24 + c1];
    acc0 = __builtin_amdgcn_wmma_f32_16x16x4_f32(false, a, false, b0,
                                                 (short)0, acc0, false, false);
    acc1 = __builtin_amdgcn_wmma_f32_16x16x4_f32(false, a, false, b1,
                                                 (short)0, acc1, false, false);
  }
}

// One wave per chunk of 16 type-sorted elements.
__global__ __launch_bounds__(128) void k_wmma(
    const float* __restrict__ U, const int* __restrict__ nodIdx,
    const float* __restrict__ filters, const int* __restrict__ types,
    const int* __restrict__ sorted, float* __restrict__ out) {
  __shared__ float sUe[4][16][25];    // padded to dodge bank conflicts
  __shared__ int   sDof[4][16][25];
  __shared__ int   sType[4][16];
  __shared__ float sKe[4][576];       // TDM-staged 24x24 filter tile per wave

  const int wib  = threadIdx.x >> 5;
  const int lane = threadIdx.x & 31;
  const int base = (blockIdx.x * 4 + wib) * 16;

  if (lane < 16) sType[wib][lane] = types[sorted[base + lane]];

  // ---- Kick the TDM for the chunk's dominant type's Ke tile (overlaps gather).
  // D# per CDNA5 ISA ch.8: count=1, type=2(image), data_size=4B,
  // tensor 24x24 == tile 24x24, dim0 stride 24 elements.
  int t0 = __builtin_amdgcn_readfirstlane(sType[wib][0]);
  {
    unsigned ldsOff = (unsigned)(uintptr_t)(&sKe[wib][0]);
    ldsOff = (unsigned)__builtin_amdgcn_readfirstlane((int)ldsOff);
    unsigned long long ga = (unsigned long long)(uintptr_t)(filters + t0 * 576);
    u32x4 g0;
    g0.x = 1u;                                      // count=1, user descriptor
    g0.y = ldsOff;                                  // lds_addr
    g0.z = (unsigned)(ga & 0xFFFFFFFFull);          // global_addr[31:0]
    g0.w = (unsigned)((ga >> 32) & 0x01FFFFFFull)   // global_addr[56:32]
         | (2u << 30);                              // type=2
    i32x8 g1;
    g1[0] = (2 << 16);          // workgroup_mask=0, data_size=2 (4 bytes)
    g1[1] = (24 << 16);         // tensor_dim0 = 24 (low 16 bits at [63:48])
    g1[2] = (24 << 16);         // tensor_dim0_hi=0 | tensor_dim1 = 24
    g1[3] = (24 << 16);         // tensor_dim1_hi=0 | tile_dim0 = 24
    g1[4] = 24;                 // tile_dim1 = 24, tile_dim2 = 0
    g1[5] = 24;                 // tensor_dim0_stride = 24
    g1[6] = 0;
    g1[7] = 0;
    i32x4 zz4 = {0, 0, 0, 0};
    i32x8 zz8 = {0, 0, 0, 0, 0, 0, 0, 0};
    // clang-23 / therock-10.0 form: 6 args (g0, g1, g2, g3, g4, cpol)
    __builtin_amdgcn_tensor_load_to_lds(g0, g1, zz4, zz4, zz8, 0);
  }

  // ---- Stage Ue (16x24 gather from U) and the scatter dof indices into LDS.
  for (int idx = lane; idx < 16 * 24; idx += 32) {
    int m = idx / 24;
    int c = idx - m * 24;
    int e = sorted[base + m];
    int node = nodIdx[e * 8 + c / 3];
    int dof = node * 3 + (c % 3);
    sDof[wib][m][c] = dof;
    sUe[wib][m][c]  = U[dof];
  }
  __syncthreads();

  const int mrow = lane & 15;   // A-matrix M index = lane%16 (ISA 16x4 f32 layout)
  const int half = lane >> 4;   // 0: K0/K1 pair, 1: K2/K3 pair
  const float* ue = &sUe[wib][mrow][0];
  v8f acc0 = {};                // output columns 0..15
  v8f acc1 = {};                // output columns 16..23 (upper 8 wasted)

  unsigned rem = 0xFFFFu;       // uniform across the wave
  bool firstPass = true;
  while (rem) {
    int fidx = __ffs(rem) - 1;
    int t = sType[wib][fidx];
    unsigned match = 0;
#pragma unroll
    for (int m = 0; m < 16; ++m) match |= (sType[wib][m] == t) ? (1u << m) : 0u;
    rem &= ~match;

    // zero-mask A rows whose element has a different type (C passes through);
    // EXEC stays all-ones as WMMA requires.
    float amask = ((match >> mrow) & 1u) ? 1.0f : 0.0f;

    if (firstPass) {
      firstPass = false;
      __builtin_amdgcn_s_wait_tensorcnt(0);           // TDM tile landed in LDS
      run_wmma_12(&sKe[wib][0], ue, amask, mrow, half, acc0, acc1);
    } else {
      // rare (~1.6% of chunks): extra distinct type, read Ke through cache
      const float* Ke = filters + t * 576;
      __builtin_prefetch(Ke, 0, 3);
      run_wmma_12(Ke, ue, amask, mrow, half, acc0, acc1);
    }
  }

  // D layout: VGPR r -> (M=r, N=lane) lanes 0-15, (M=r+8, N=lane-16) lanes 16-31
#pragma unroll
  for (int r = 0; r < 8; ++r) {
    int m = r + half * 8;
    atomicAdd(&out[sDof[wib][m][mrow]], acc0[r]);
    if (mrow < 8) atomicAdd(&out[sDof[wib][m][16 + mrow]], acc1[r]);
  }
}

extern "C" void kernel_launch(void* const* d_in, const int* in_sizes, int n_in,
                              void* d_out, int out_size, void* d_ws, size_t ws_size,
                              hipStream_t stream) {
  const float* U       = (const float*)d_in[0];
  const float* rho     = (const float*)d_in[1];
  const int*   nodIdx  = (const int*)d_in[2];
  const float* filters = (const float*)d_in[3];
  // d_in[4] (typeFilter) is the fixed 2^(di*4+dj*2+dk) kernel; baked into k_types.
  (void)in_sizes; (void)n_in; (void)out_size; (void)ws_size;

  float* out = (float*)d_out;
  int* ws      = (int*)d_ws;
  int* types   = ws;
  int* sorted  = ws + NEL;
  int* counts  = ws + 2 * NEL;
  int* cursors = ws + 2 * NEL + 256;

  k_init   <<<(NDOF + 255) / 256, 256, 0, stream>>>(out, counts);
  k_types  <<<(NEL  + 255) / 256, 256, 0, stream>>>(rho, types, counts);
  k_scan   <<<1, 256, 0, stream>>>(counts, cursors);
  k_scatter<<<(NEL  + 255) / 256, 256, 0, stream>>>(types, cursors, sorted);
  k_wmma   <<<NCHUNK / 4, 128, 0, stream>>>(U, nodIdx, filters, types, sorted, out);
}